// CDTripletLoss_73813307949745
// MI455X (gfx1250) — compile-verified
//
#include <hip/hip_runtime.h>
#include <hip/hip_bf16.h>

#define C_COMM 512
#define D_DIM  128
#define CD     (C_COMM * D_DIM)
#define ALPHA_C 0.25f
#define P_EPS_C 1e-6f
#define BIG_C   1e30f
#define LDS_STRIDE 130   // 128 + 2 pad -> conflict-free ds_load_b64

typedef __attribute__((ext_vector_type(2))) float v2f;
typedef __attribute__((ext_vector_type(8))) float v8f;

// ---------------------------------------------------------------------------
// async copy of 32B (2 x b128) global -> LDS, tracked with ASYNCcnt.
// INST_OFFSET applies to both LDS and global addresses (CDNA5 ISA)
// ---------------------------------------------------------------------------
__device__ __forceinline__ void stage32B_async(const float* __restrict__ gsrc,
                                               float* ldst) {
    unsigned ldsa = (unsigned)(size_t)ldst;   // low 32 bits of flat addr == LDS offset
    asm volatile(
        "global_load_async_to_lds_b128 %0, %1, off\n\t"
        "global_load_async_to_lds_b128 %0, %1, off offset:16"
        :: "v"(ldsa), "v"(gsrc)
        : "memory");
}

__device__ __forceinline__ void wait_async_all() {
    asm volatile("s_wait_asynccnt 0x0" ::: "memory");
}

// ---------------------------------------------------------------------------
// K0: zero the accumulated regions of the workspace (comm_sum, sq, counts, acc)
// ---------------------------------------------------------------------------
__global__ void zero_kernel(float* __restrict__ p, int n) {
    int i = blockIdx.x * blockDim.x + threadIdx.x;
    if (i < n) p[i] = 0.0f;
}

// ---------------------------------------------------------------------------
// K1: comm[node] scatter.  cbl row0 = perm, row1 = community (perm order).
// ---------------------------------------------------------------------------
__global__ void scatter_comm_kernel(const int* __restrict__ cbl,
                                    int* __restrict__ comm, int N) {
    int i = blockIdx.x * blockDim.x + threadIdx.x;
    if (i < N) comm[cbl[i]] = cbl[N + i];
}

// ---------------------------------------------------------------------------
// K2: counts (atomic) + per-node squared norm x2[n]
// ---------------------------------------------------------------------------
__global__ void count_x2_kernel(const float* __restrict__ X,
                                const int* __restrict__ comm,
                                float* __restrict__ counts,
                                float* __restrict__ x2, int N) {
    int n = blockIdx.x * blockDim.x + threadIdx.x;
    if (n >= N) return;
    atomicAdd(&counts[comm[n]], 1.0f);
    const float4* xp = (const float4*)(X + (size_t)n * D_DIM);
    float s = 0.0f;
#pragma unroll 8
    for (int j = 0; j < D_DIM / 4; ++j) {
        float4 x = xp[j];
        s += x.x * x.x + x.y * x.y + x.z * x.z + x.w * x.w;
    }
    x2[n] = s;
}

// ---------------------------------------------------------------------------
// K3: comm_sum segment reduction (atomics into 256KB L2-resident buffer)
// ---------------------------------------------------------------------------
__global__ void accum_sum_kernel(const float* __restrict__ X,
                                 const int* __restrict__ comm,
                                 float* __restrict__ comm_sum, int N) {
    int idx = blockIdx.x * blockDim.x + threadIdx.x;
    if (idx >= N * (D_DIM / 4)) return;
    int n  = idx >> 5;            // D_DIM/4 == 32
    int d4 = (idx & 31) << 2;
    int c  = comm[n];
    float4 x = *(const float4*)(X + (size_t)n * D_DIM + d4);
    float* dst = comm_sum + (size_t)c * D_DIM + d4;
    atomicAdd(dst + 0, x.x);
    atomicAdd(dst + 1, x.y);
    atomicAdd(dst + 2, x.z);
    atomicAdd(dst + 3, x.w);
}

// ---------------------------------------------------------------------------
// K4: mean = comm_sum / counts ; c2[c] = ||mean_c||^2   (one block / community)
// ---------------------------------------------------------------------------
__global__ __launch_bounds__(128) void mean_c2_kernel(
        const float* __restrict__ comm_sum, const float* __restrict__ counts,
        float* __restrict__ meanp, float* __restrict__ c2) {
    int c = blockIdx.x;
    int d = threadIdx.x;
    float cnt = counts[c];
    float m = comm_sum[(size_t)c * D_DIM + d] / fmaxf(cnt, 1.0f);
    meanp[(size_t)c * D_DIM + d] = m;
    float v = m * m;
#pragma unroll
    for (int msk = 1; msk < 32; msk <<= 1) v += __shfl_xor(v, msk, 32);
    __shared__ float ls[4];
    if ((threadIdx.x & 31) == 0) ls[threadIdx.x >> 5] = v;
    __syncthreads();
    if (threadIdx.x == 0) c2[c] = ls[0] + ls[1] + ls[2] + ls[3];
}

// ---------------------------------------------------------------------------
// K5: fused WMMA distance kernel, LDS-staged B with async double-buffering.
// One wave32 per 16-node tile, 8 tiles per block.  A fragments (16x4 f32,
// K=128) live in registers; the 16-community mean chunk for each c-tile is
// staged global->LDS with GLOBAL_LOAD_ASYNC_TO_LDS_B128 and shared by all
// 8 waves (8x less L2 traffic, DS-latency B fragments).
// ---------------------------------------------------------------------------
__global__ __launch_bounds__(256) void dist_wmma_kernel(
        const float* __restrict__ X, const float* __restrict__ meanp,
        const float* __restrict__ x2, const float* __restrict__ c2,
        const int* __restrict__ comm,
        float* __restrict__ rowSum, float* __restrict__ rowMin,
        float* __restrict__ rowOwn, int N) {
    __shared__ float smem[2][16 * LDS_STRIDE];

    const int wave = threadIdx.x >> 5;
    const int lane = threadIdx.x & 31;
    const int tile = blockIdx.x * 8 + wave;
    int n0 = tile * 16;
    const bool active = (n0 < N);
    if (!active) n0 = ((N - 1) >> 4) << 4;   // clamp: reads only, stores guarded

    const int half = lane >> 4;              // 0 or 1
    const int l16  = lane & 15;

    // per-thread staging slot: row = tid/16, 8-float segment = (tid%16)*8
    const int srow = threadIdx.x >> 4;
    const int sseg = (threadIdx.x & 15) << 3;

    // stage community chunk 0
    stage32B_async(meanp + (size_t)srow * D_DIM + sseg,
                   &smem[0][srow * LDS_STRIDE + sseg]);

    // --- A fragments: row M = n0+l16, K = 4*ks + 2*half + {0,1} (ISA layout)
    int nrow = n0 + l16; if (nrow > N - 1) nrow = N - 1;
    const float* xp = X + (size_t)nrow * D_DIM + 2 * half;
    v2f af[32];
#pragma unroll
    for (int ks = 0; ks < 32; ++ks)
        af[ks] = *(const v2f*)(xp + ks * 4);

    // --- per-output-row constants (D layout rows: v + 8*half)
    float xr[8];
    int   cr[8];
#pragma unroll
    for (int v = 0; v < 8; ++v) {
        int r = n0 + v + 8 * half; if (r > N - 1) r = N - 1;
        xr[v] = x2[r];
        cr[v] = comm[r];
    }

    float sumA[8], minA[8], ownA[8];
#pragma unroll
    for (int v = 0; v < 8; ++v) { sumA[v] = 0.0f; minA[v] = BIG_C; ownA[v] = 0.0f; }

    wait_async_all();
    __syncthreads();

    for (int ct = 0; ct < C_COMM / 16; ++ct) {
        const int cur = ct & 1;
        // prefetch next chunk while computing on current one
        if (ct + 1 < C_COMM / 16) {
            stage32B_async(meanp + (size_t)((ct + 1) * 16 + srow) * D_DIM + sseg,
                           &smem[cur ^ 1][srow * LDS_STRIDE + sseg]);
        }

        const int c0 = ct * 16;
        v8f acc = {};
        const float* bp = &smem[cur][l16 * LDS_STRIDE + 2 * half];
#pragma unroll
        for (int g = 0; g < 4; ++g) {
            v2f bf[8];
#pragma unroll
            for (int k = 0; k < 8; ++k)
                bf[k] = *(const v2f*)(bp + (g * 8 + k) * 4);
#pragma unroll
            for (int k = 0; k < 8; ++k)
                acc = __builtin_amdgcn_wmma_f32_16x16x4_f32(
                        false, af[g * 8 + k], false, bf[k], (short)0, acc,
                        false, false);
        }

        const float c2v  = c2[c0 + l16];
        const int   ccol = c0 + l16;
#pragma unroll
        for (int v = 0; v < 8; ++v) {
            float d2   = xr[v] + c2v - 2.0f * acc[v];
            float dist = sqrtf(fmaxf(d2, 0.0f));
            bool  own  = (cr[v] == ccol);
            sumA[v] += dist;
            ownA[v] += own ? dist : 0.0f;
            minA[v]  = fminf(minA[v], own ? BIG_C : dist);
        }

        if (ct + 1 < C_COMM / 16) {
            wait_async_all();     // next chunk landed in LDS
            __syncthreads();      // all waves done reading / staging
        }
    }

    // reduce across the 16 lanes of each half (columns of each row)
#pragma unroll
    for (int v = 0; v < 8; ++v) {
#pragma unroll
        for (int m = 1; m < 16; m <<= 1) {
            sumA[v] += __shfl_xor(sumA[v], m, 16);
            ownA[v] += __shfl_xor(ownA[v], m, 16);
            minA[v]  = fminf(minA[v], __shfl_xor(minA[v], m, 16));
        }
    }
    if (active && l16 == 0) {
#pragma unroll
        for (int v = 0; v < 8; ++v) {
            int r = n0 + v + 8 * half;
            if (r < N) { rowSum[r] = sumA[v]; rowMin[r] = minA[v]; rowOwn[r] = ownA[v]; }
        }
    }
}

// ---------------------------------------------------------------------------
// K6: sq[c][d] += (x - mean[c][d])^2   (atomics, one thread per (n, d4))
// ---------------------------------------------------------------------------
__global__ void sq_kernel(const float* __restrict__ X,
                          const float* __restrict__ meanp,
                          const int* __restrict__ comm,
                          float* __restrict__ sq, int N) {
    int idx = blockIdx.x * blockDim.x + threadIdx.x;
    if (idx >= N * (D_DIM / 4)) return;
    int n  = idx >> 5;
    int d4 = (idx & 31) << 2;
    int c  = comm[n];
    float4 x = *(const float4*)(X + (size_t)n * D_DIM + d4);
    float4 m = *(const float4*)(meanp + (size_t)c * D_DIM + d4);
    float* dst = sq + (size_t)c * D_DIM + d4;
    float dx = x.x - m.x, dy = x.y - m.y, dz = x.z - m.z, dw = x.w - m.w;
    atomicAdd(dst + 0, dx * dx);
    atomicAdd(dst + 1, dy * dy);
    atomicAdd(dst + 2, dz * dz);
    atomicAdd(dst + 3, dw * dw);
}

// ---------------------------------------------------------------------------
// K7: pos_dist + triplet terms, one wave per node; accumulate into acc[0..1]
// ---------------------------------------------------------------------------
__global__ __launch_bounds__(256) void pos_trip_kernel(
        const float* __restrict__ X, const float* __restrict__ comm_sum,
        const float* __restrict__ counts, const int* __restrict__ comm,
        const float* __restrict__ rowSum, const float* __restrict__ rowMin,
        const float* __restrict__ rowOwn, float* __restrict__ acc, int N) {
    int wave = threadIdx.x >> 5;
    int lane = threadIdx.x & 31;
    int n = blockIdx.x * 8 + wave;
    if (n >= N) return;
    int   c    = comm[n];
    float cnt  = counts[c];
    float cm1  = fmaxf(cnt - 1.0f, 1.0f);
    float xsc  = (cnt == 1.0f) ? 0.0f : 1.0f;   // x_minus masking
    float4 x = *(const float4*)(X + (size_t)n * D_DIM + lane * 4);
    float4 s = *(const float4*)(comm_sum + (size_t)c * D_DIM + lane * 4);
    float p = 0.0f;
    {
        float t;
        t = (s.x - x.x * xsc) / cm1 - x.x + P_EPS_C; p += t * t;
        t = (s.y - x.y * xsc) / cm1 - x.y + P_EPS_C; p += t * t;
        t = (s.z - x.z * xsc) / cm1 - x.z + P_EPS_C; p += t * t;
        t = (s.w - x.w * xsc) / cm1 - x.w + P_EPS_C; p += t * t;
    }
#pragma unroll
    for (int m = 1; m < 32; m <<= 1) p += __shfl_xor(p, m, 32);
    if (lane == 0) {
        float pos      = sqrtf(p);
        float mean_neg = (rowSum[n] - rowOwn[n]) * (1.0f / (C_COMM - 1));
        float tm = fmaxf(pos - mean_neg + ALPHA_C, 0.0f);
        float ti = fmaxf(pos - rowMin[n] + ALPHA_C, 0.0f);
        atomicAdd(&acc[0], tm);
        atomicAdd(&acc[1], ti);
    }
}

// ---------------------------------------------------------------------------
// K8: std loss term, one thread per (c,d); wave-reduce then atomic
// ---------------------------------------------------------------------------
__global__ void std_kernel(const float* __restrict__ sq,
                           const float* __restrict__ counts,
                           float* __restrict__ acc) {
    int idx = blockIdx.x * blockDim.x + threadIdx.x;   // CD threads exactly
    int c = idx >> 7;
    float cm1 = fmaxf(counts[c] - 1.0f, 1.0f);
    float s = sqrtf(sq[idx] / cm1) - 1.0f;             // STD_TARGET = 1
    float v = s * s;
#pragma unroll
    for (int m = 1; m < 32; m <<= 1) v += __shfl_xor(v, m, 32);
    if ((threadIdx.x & 31) == 0) atomicAdd(&acc[2], v);
}

// ---------------------------------------------------------------------------
// K9: finalize the three scalar outputs
// ---------------------------------------------------------------------------
__global__ void finalize_kernel(const float* __restrict__ acc,
                                float* __restrict__ out, int N) {
    if (threadIdx.x == 0 && blockIdx.x == 0) {
        out[0] = acc[0] / (float)N;
        out[1] = acc[1] / (float)N;
        out[2] = acc[2] / (float)CD;
    }
}

// ---------------------------------------------------------------------------
extern "C" void kernel_launch(void* const* d_in, const int* in_sizes, int n_in,
                              void* d_out, int out_size, void* d_ws, size_t ws_size,
                              hipStream_t stream) {
    const float* X   = (const float*)d_in[0];
    const int*   cbl = (const int*)d_in[1];
    float* out = (float*)d_out;

    const int N = in_sizes[1] / 2;          // community_belong_list is (2, N)

    // ---- workspace layout (floats) ----
    float* ws       = (float*)d_ws;
    float* comm_sum = ws;                   // CD
    float* sq       = ws + CD;              // CD
    float* counts   = ws + 2 * CD;          // C
    float* acc      = ws + 2 * CD + C_COMM; // 16 (padded)
    const int ZERO_N = 2 * CD + C_COMM + 16;
    float* c2     = ws + ZERO_N;                 // C
    float* meanp  = c2 + C_COMM;                 // CD
    float* x2     = meanp + CD;                  // N
    float* rowSum = x2 + N;                      // N
    float* rowMin = rowSum + N;                  // N
    float* rowOwn = rowMin + N;                  // N
    int*   comm   = (int*)(rowOwn + N);          // N ints

    const int T = 256;
    zero_kernel<<<(ZERO_N + T - 1) / T, T, 0, stream>>>(ws, ZERO_N);
    scatter_comm_kernel<<<(N + T - 1) / T, T, 0, stream>>>(cbl, comm, N);
    count_x2_kernel<<<(N + T - 1) / T, T, 0, stream>>>(X, comm, counts, x2, N);
    accum_sum_kernel<<<(N * (D_DIM / 4) + T - 1) / T, T, 0, stream>>>(X, comm, comm_sum, N);
    mean_c2_kernel<<<C_COMM, 128, 0, stream>>>(comm_sum, counts, meanp, c2);

    const int tiles = (N + 15) / 16;
    dist_wmma_kernel<<<(tiles + 7) / 8, T, 0, stream>>>(
        X, meanp, x2, c2, comm, rowSum, rowMin, rowOwn, N);

    sq_kernel<<<(N * (D_DIM / 4) + T - 1) / T, T, 0, stream>>>(X, meanp, comm, sq, N);
    pos_trip_kernel<<<(N + 7) / 8, T, 0, stream>>>(
        X, comm_sum, counts, comm, rowSum, rowMin, rowOwn, acc, N);
    std_kernel<<<CD / T, T, 0, stream>>>(sq, counts, acc);
    finalize_kernel<<<1, 1, 0, stream>>>(acc, out, N);
}